// Graph_Learn_3796751090346
// MI455X (gfx1250) — compile-verified
//
#include <hip/hip_runtime.h>
#include <hip/hip_bf16.h>

#define N_BATCH 32
#define V 256
#define F 64
#define XPAD 65           // padded LDS row stride (floats) -> conflict-free column reads
#define ALPHA 1e-4f

typedef __attribute__((ext_vector_type(2))) float v2f;
typedef __attribute__((ext_vector_type(8))) float v8f;

// One workgroup per batch n. Thread t owns column j = t.
__global__ __launch_bounds__(256) void graph_learn_main(
    const float* __restrict__ x,      // (N, V, F)
    const float* __restrict__ a,      // (F)
    float* __restrict__ out,          // (N, V, V) S, then 1 loss scalar (written by finalize)
    float* __restrict__ partials)     // ws: [0..31] = Sloss part, [32..63] = dloss part
{
    __shared__ float sx[V * XPAD];    // X[n], padded            (~65 KB)
    __shared__ float sa[F];           // a
    __shared__ float sq[V];           // q_i = ||x_i||^2
    __shared__ float sG[16 * V];      // 16x256 Gram strip       (16 KB)
    __shared__ float sred[2 * 256];   // loss reduction scratch

    const int n = blockIdx.x;
    const int t = threadIdx.x;        // column j
    const float* xn = x + (size_t)n * V * F;
    float* outn = out + (size_t)n * V * V;

    if (t < F) sa[t] = a[t];
    // Coalesced load of X[n] into padded LDS
    for (int idx = t; idx < V * F; idx += 256) {
        int r = idx >> 6, c = idx & 63;
        sx[r * XPAD + c] = xn[idx];
    }
    __syncthreads();

    // Own column -> registers (fully unrolled so xj[] stays in VGPRs), plus q_j
    float xj[F];
    float qj = 0.f;
    #pragma unroll
    for (int f = 0; f < F; ++f) {
        float v = sx[t * XPAD + f];
        xj[f] = v;
        qj = fmaf(v, v, qj);
    }
    sq[t] = qj;
    __syncthreads();

    const int wave  = t >> 5;
    const int lane  = t & 31;
    const int lhalf = lane >> 4;      // 0: lanes 0-15, 1: lanes 16-31
    const int l15   = lane & 15;

    float colsum = 0.f;               // sum_i tmpS[i, j]
    float snum   = 0.f;               // sum_i tmpS^2
    float dnum   = 0.f;               // sum_i d2 * tmpS

    for (int i0 = 0; i0 < V; i0 += 16) {
        // ---- WMMA: Gram strip G[i0..i0+15, 0..255], K = 64, f32 in / f32 out ----
        // 8 waves x 2 column-tiles each = 16 tiles of 16x16.
        #pragma unroll
        for (int tt = 0; tt < 2; ++tt) {
            const int j0 = (wave * 2 + tt) * 16;
            v8f acc = {};
            #pragma unroll
            for (int kk = 0; kk < F; kk += 4) {
                const int ko = kk + 2 * lhalf;
                // A 16x4 (M=i0+l15, K=ko/ko+1), B 4x16 (N=j0+l15, same K split)
                v2f A, B;
                A.x = sx[(i0 + l15) * XPAD + ko];
                A.y = sx[(i0 + l15) * XPAD + ko + 1];
                B.x = sx[(j0 + l15) * XPAD + ko];
                B.y = sx[(j0 + l15) * XPAD + ko + 1];
                acc = __builtin_amdgcn_wmma_f32_16x16x4_f32(
                    false, A, false, B, (short)0, acc, false, false);
            }
            // C/D layout: VGPR r, lanes 0-15 -> M=r, lanes 16-31 -> M=r+8
            #pragma unroll
            for (int r = 0; r < 8; ++r) {
                sG[(r + 8 * lhalf) * V + j0 + l15] = acc[r];
            }
        }
        __syncthreads();

        // ---- scalar abs-weighted part for rows i0..i0+15 ----
        for (int ii = 0; ii < 16; ++ii) {
            const int i = i0 + ii;
            float s1 = 0.f;
            #pragma unroll
            for (int f = 0; f < F; ++f) {
                float d = sx[i * XPAD + f] - xj[f];   // broadcast LDS read + register
                s1 = fmaf(sa[f], fabsf(d), s1);
            }
            const float tmpS = __expf(-fmaxf(s1, 0.f));
            colsum += tmpS;
            const float d2 = sq[i] + qj - 2.f * sG[ii * V + t];
            dnum = fmaf(d2, tmpS, dnum);
            snum = fmaf(tmpS, tmpS, snum);
            outn[i * V + t] = tmpS;                   // raw tmpS, coalesced
        }
        __syncthreads();  // sG reused next strip
    }

    // Deferred normalization: rescale own column of d_out (same-thread RMW)
    const float rinv = 1.0f / colsum;
    for (int i = 0; i < V; ++i) {
        outn[i * V + t] *= rinv;
    }

    // Per-thread loss contributions, divide deferred to here
    const float lS = snum * rinv * rinv;   // sum_i S^2 for column j
    const float lD = dnum * rinv;          // sum_i d2*S for column j

    sred[t]       = lS;
    sred[256 + t] = lD;
    __syncthreads();
    for (int s = 128; s > 0; s >>= 1) {
        if (t < s) {
            sred[t]       += sred[t + s];
            sred[256 + t] += sred[256 + t + s];
        }
        __syncthreads();
    }
    if (t == 0) {
        partials[n]            = sred[0];
        partials[N_BATCH + n]  = sred[256];
    }
}

// Fixed-order summation -> bitwise-deterministic loss across graph replays
__global__ void graph_learn_finalize(const float* __restrict__ partials,
                                     float* __restrict__ out_loss)
{
    float sS = 0.f, sD = 0.f;
    for (int n = 0; n < N_BATCH; ++n) {
        sS += partials[n];
        sD += partials[N_BATCH + n];
    }
    *out_loss = (sS + ALPHA * sD) * (1.0f / (float)N_BATCH);
}

extern "C" void kernel_launch(void* const* d_in, const int* in_sizes, int n_in,
                              void* d_out, int out_size, void* d_ws, size_t ws_size,
                              hipStream_t stream) {
    const float* x = (const float*)d_in[0];   // (32, 256, 64) f32
    const float* a = (const float*)d_in[1];   // (64, 1) f32
    float* out      = (float*)d_out;          // 32*256*256 S values + 1 loss
    float* partials = (float*)d_ws;           // 64 floats

    graph_learn_main<<<dim3(N_BATCH), dim3(256), 0, stream>>>(x, a, out, partials);
    graph_learn_finalize<<<1, 1, 0, stream>>>(partials, out + (size_t)N_BATCH * V * V);
}